// DetectionLoss_6253472383768
// MI455X (gfx1250) — compile-verified
//
#include <hip/hip_runtime.h>
#include <hip/hip_bf16.h>
#include <stdint.h>

// Problem constants (from reference): B=8, A=65536, O=64, C=81
#define BB 8
#define AA 65536
#define OO 64
#define CC 81
#define RB 128           // rows (anchors) per CE block
#define POS_CAP 131072   // capacity of compact positive list

// ---------------------------------------------------------------------------
// CDNA5 async-tensor data path: global -> LDS async copies (ASYNCcnt tracked)
// ---------------------------------------------------------------------------
__device__ __forceinline__ uint32_t lds_off_of(const void* p) {
  // LDS aperture: addr[63:32] = SHARED_BASE, addr[31:0] = wave LDS byte offset
  return (uint32_t)(uintptr_t)p;
}
__device__ __forceinline__ void async_ld_b32(uint32_t lds_off, const void* g) {
  asm volatile("global_load_async_to_lds_b32 %0, %1, off"
               :: "v"(lds_off), "v"(g) : "memory");
}
__device__ __forceinline__ void async_ld_b128(uint32_t lds_off, const void* g) {
  asm volatile("global_load_async_to_lds_b128 %0, %1, off"
               :: "v"(lds_off), "v"(g) : "memory");
}
__device__ __forceinline__ void wait_async0() {
  asm volatile("s_wait_asynccnt 0" ::: "memory");
}

// ---------------------------------------------------------------------------
// helpers
// ---------------------------------------------------------------------------
__device__ __forceinline__ float iou1(float4 a, const float* bp) {
  float bx1 = bp[0], by1 = bp[1], bx2 = bp[2], by2 = bp[3];
  float lx = fmaxf(a.x, bx1), ly = fmaxf(a.y, by1);
  float rx = fminf(a.z, bx2), ry = fminf(a.w, by2);
  float w = fmaxf(rx - lx, 0.0f), h = fmaxf(ry - ly, 0.0f);
  float inter = w * h;
  float ar_a = (a.z - a.x) * (a.w - a.y);
  float ar_b = (bx2 - bx1) * (by2 - by1);
  return inter / (ar_a + ar_b - inter);
}
__device__ __forceinline__ float sl1(float d) {
  d = fabsf(d);
  return (d < 1.0f) ? 0.5f * d * d : d - 0.5f;
}
__device__ __forceinline__ float n2n(float x) {  // jnp.nan_to_num defaults
  if (isnan(x)) return 0.0f;
  if (isinf(x)) return (x > 0.0f) ? 3.4028234663852886e38f : -3.4028234663852886e38f;
  return x;
}

// ---------------------------------------------------------------------------
// K0: zero counters / accumulators / histogram (every launch: replay-safe)
// ---------------------------------------------------------------------------
__global__ void kInit(int* cnt, float* acc, int* hist) {
  int t = threadIdx.x;
  if (t < 16) { cnt[t] = 0; acc[t] = 0.0f; }
  hist[t] = 0;  // blockDim.x == 256
}

// ---------------------------------------------------------------------------
// K1: anchor<->gt matching. gt boxes + labels broadcast to LDS via async DMA.
// ---------------------------------------------------------------------------
__global__ void kMatch(const float* __restrict__ anchors,
                       const float* __restrict__ gt,
                       const int* __restrict__ labels,
                       float* __restrict__ mo_out,
                       int* __restrict__ cnt, int2* __restrict__ pos) {
  __shared__ __align__(16) float sgt[OO * 4];   // 1 KB
  __shared__ int slab[OO];
  const int t = threadIdx.x;
  const int b = blockIdx.y;
  const int a = blockIdx.x * 256 + t;

  async_ld_b32(lds_off_of(&sgt[t]), gt + (size_t)b * OO * 4 + t);       // 256 floats
  if (t < OO)
    async_ld_b32(lds_off_of(&slab[t]), labels + b * OO + t);            // 64 ints
  wait_async0();
  __syncthreads();

  float4 av = reinterpret_cast<const float4*>(anchors)[a];
  float mo = -1.0f;
#pragma unroll 4
  for (int o = 0; o < OO; ++o) {
    float ov = (slab[o] < 0) ? -1.0f : iou1(av, &sgt[o * 4]);
    mo = fmaxf(mo, ov);
  }
  mo_out[(size_t)b * AA + a] = mo;

  if (mo > 0.5f) {  // rare: second pass reproduces |mo-ov|<1e-6 tie semantics
    for (int o = 0; o < OO; ++o) {
      float ov = (slab[o] < 0) ? -1.0f : iou1(av, &sgt[o * 4]);
      if (ov > 0.5f && fabsf(mo - ov) < 1e-6f) {
        int idx = atomicAdd(&cnt[0], 1);
        if (idx < POS_CAP) { pos[idx].x = b * AA + a; pos[idx].y = o; }
      }
    }
  }
}

// ---------------------------------------------------------------------------
// K2: streaming log-sum-exp over 170MB of pred_classes.
// Tile (128 rows x 81 cols = 41472B) staged into LDS with async b128 copies:
// fully coalesced 16B/lane; LDS row stride 81 dwords is coprime with 64 banks.
// ---------------------------------------------------------------------------
__global__ void kCE(const float* __restrict__ pc,
                    const float* __restrict__ mo_in,
                    float* __restrict__ lse_out,
                    float* __restrict__ ceneg,
                    int* __restrict__ cnt) {
  __shared__ __align__(16) float tile[RB * CC];
  const int t = threadIdx.x;
  const int b = blockIdx.y;
  const int a0 = blockIdx.x * RB;
  const float* src = pc + ((size_t)b * AA + a0) * CC;
  const uint32_t base = lds_off_of(tile);

  for (int i = t; i < (RB * CC) / 4; i += RB)                 // 2592 x 16B chunks
    async_ld_b128(base + (uint32_t)i * 16u, src + (size_t)i * 4);
  wait_async0();
  __syncthreads();

  const float* row = &tile[t * CC];
  float m = row[0];
#pragma unroll
  for (int j = 1; j < CC; ++j) m = fmaxf(m, row[j]);
  float s = 0.0f;
#pragma unroll
  for (int j = 0; j < CC; ++j) s += __expf(row[j] - m);
  float lse = m + __logf(s);

  int ga = b * AA + a0 + t;
  lse_out[ga] = lse;
  if (mo_in[ga] < 0.5f) {                        // negative: target class 0
    int idx = atomicAdd(&cnt[1], 1);
    ceneg[idx] = lse - row[0];                   // ce >= 0 always
  }
}

// ---------------------------------------------------------------------------
// K3: positives — SSD-encode + smooth-L1 sum and positive CE
// ---------------------------------------------------------------------------
__global__ void kPos(const float* __restrict__ pb, const float* __restrict__ pc,
                     const float* __restrict__ gt, const float* __restrict__ anchors,
                     const int* __restrict__ labels, const float* __restrict__ lse_in,
                     const int* __restrict__ cnt, const int2* __restrict__ pos,
                     float* __restrict__ acc) {
  int P = cnt[0]; if (P > POS_CAP) P = POS_CAP;
  for (int i = blockIdx.x * blockDim.x + threadIdx.x; i < P;
       i += gridDim.x * blockDim.x) {
    int ba = pos[i].x, o = pos[i].y;
    int b = ba >> 16;                 // A == 65536
    int a = ba & (AA - 1);
    float4 g  = reinterpret_cast<const float4*>(gt)[b * OO + o];
    float4 an = reinterpret_cast<const float4*>(anchors)[a];
    float4 p  = reinterpret_cast<const float4*>(pb)[ba];
    float aw = an.z - an.x, ah = an.w - an.y;
    float tx = (0.5f * (g.x + g.z) - 0.5f * (an.x + an.z)) / (0.1f * aw);
    float ty = (0.5f * (g.y + g.w) - 0.5f * (an.y + an.w)) / (0.1f * ah);
    float tw = __logf((g.z - g.x) / aw) / 0.2f;
    float th = __logf((g.w - g.y) / ah) / 0.2f;
    float l = sl1(p.x - tx) + sl1(p.y - ty) + sl1(p.z - tw) + sl1(p.w - th);
    atomicAdd(&acc[0], l);
    int lab = labels[b * OO + o];
    float ce = lse_in[ba] - pc[(size_t)ba * CC + lab];
    atomicAdd(&acc[1], ce);
  }
}

// ---------------------------------------------------------------------------
// K4: k = min(10*P, Nneg); reset radix-select state
// ---------------------------------------------------------------------------
__global__ void kSetK(int* cnt) {
  long long k = (long long)cnt[0] * 10;
  if (k > cnt[1]) k = cnt[1];
  cnt[2] = (int)k;   // remaining count during select
  cnt[3] = 0;        // fixed-prefix bits (right-aligned)
  cnt[4] = (int)k;   // original k
}

// ---------------------------------------------------------------------------
// K5a/b: exact radix top-k select on CE bit patterns (ce >= 0 => monotone bits)
// ---------------------------------------------------------------------------
__global__ void kHist(const int* __restrict__ cnt, int* __restrict__ hist,
                      const float* __restrict__ ceneg, int shift) {
  __shared__ int sh[256];
  int t = threadIdx.x;
  sh[t] = 0;
  __syncthreads();
  int Nn = cnt[1];
  uint32_t prefix = (uint32_t)cnt[3];
  const uint32_t* v = (const uint32_t*)ceneg;
  for (int i = blockIdx.x * blockDim.x + t; i < Nn; i += gridDim.x * blockDim.x) {
    uint32_t x = v[i];
    uint32_t hv = (shift >= 24) ? 0u : (x >> (shift + 8));
    if (hv == prefix) atomicAdd(&sh[(x >> shift) & 0xFFu], 1);
  }
  __syncthreads();
  if (sh[t]) atomicAdd(&hist[t], sh[t]);
}

__global__ void kScan(int* cnt, int* hist) {
  if (threadIdx.x == 0) {
    int rem = cnt[2];
    uint32_t prefix = (uint32_t)cnt[3];
    int d = 255;
    for (; d > 0; --d) {
      int c = hist[d];
      if (c < rem) rem -= c; else break;
    }
    cnt[3] = (int)((prefix << 8) | (uint32_t)d);
    cnt[2] = rem;   // values equal to threshold still needed
  }
  __syncthreads();
  hist[threadIdx.x] = 0;   // ready for next pass
}

// K6: sum of all ce strictly above threshold T (ties handled via rem*T)
__global__ void kSumTop(const int* __restrict__ cnt,
                        const float* __restrict__ ceneg, float* __restrict__ acc) {
  if (cnt[4] <= 0) return;
  int Nn = cnt[1];
  uint32_t T = (uint32_t)cnt[3];
  const uint32_t* v = (const uint32_t*)ceneg;
  float local = 0.0f;
  for (int i = blockIdx.x * blockDim.x + threadIdx.x; i < Nn;
       i += gridDim.x * blockDim.x)
    if (v[i] > T) local += ceneg[i];
  if (local != 0.0f) atomicAdd(&acc[2], local);
}

// K7: finalize losses (reproduce jnp.nan_to_num placement exactly)
__global__ void kFinal(const int* __restrict__ cnt, const float* __restrict__ acc,
                       float* __restrict__ out) {
  int P = cnt[0];
  int k0 = cnt[4];
  int rem = cnt[2];
  uint32_t T = (uint32_t)cnt[3];
  float topk = (k0 > 0) ? (acc[2] + (float)rem * __uint_as_float(T)) : 0.0f;
  float boxes = n2n(acc[0]) / (float)(P > 1 ? P : 1);
  int n_sel = P + k0; if (n_sel < 1) n_sel = 1;
  float mean_ce = (acc[1] + topk) / (float)n_sel;
  float classes = n2n(mean_ce) / (float)n_sel;
  out[0] = boxes;
  out[1] = classes;
  out[2] = boxes + classes;
}

// ---------------------------------------------------------------------------
extern "C" void kernel_launch(void* const* d_in, const int* in_sizes, int n_in,
                              void* d_out, int out_size, void* d_ws, size_t ws_size,
                              hipStream_t stream) {
  (void)in_sizes; (void)n_in; (void)out_size; (void)ws_size;
  const float* pred_boxes   = (const float*)d_in[0];
  const float* pred_classes = (const float*)d_in[1];
  const float* gt_boxes     = (const float*)d_in[2];
  const int*   labels       = (const int*)d_in[3];
  const float* anchors      = (const float*)d_in[4];
  float* out = (float*)d_out;

  char* ws = (char*)d_ws;
  int*   cnt   = (int*)(ws + 0);                                   // 64 B
  float* acc   = (float*)(ws + 64);                                // 64 B
  int*   hist  = (int*)(ws + 128);                                 // 1 KB
  float* mo    = (float*)(ws + 4096);                              // 2 MB
  float* lse   = (float*)(ws + 4096 + 1 * (size_t)BB * AA * 4);    // 2 MB
  float* ceneg = (float*)(ws + 4096 + 2 * (size_t)BB * AA * 4);    // 2 MB
  int2*  pos   = (int2*) (ws + 4096 + 3 * (size_t)BB * AA * 4);    // 1 MB

  kInit<<<1, 256, 0, stream>>>(cnt, acc, hist);
  kMatch<<<dim3(AA / 256, BB), 256, 0, stream>>>(anchors, gt_boxes, labels,
                                                 mo, cnt, pos);
  kCE<<<dim3(AA / RB, BB), RB, 0, stream>>>(pred_classes, mo, lse, ceneg, cnt);
  kPos<<<256, 256, 0, stream>>>(pred_boxes, pred_classes, gt_boxes, anchors,
                                labels, lse, cnt, pos, acc);
  kSetK<<<1, 1, 0, stream>>>(cnt);
  for (int shift = 24; shift >= 0; shift -= 8) {
    kHist<<<256, 256, 0, stream>>>(cnt, hist, ceneg, shift);
    kScan<<<1, 256, 0, stream>>>(cnt, hist);
  }
  kSumTop<<<256, 256, 0, stream>>>(cnt, ceneg, acc);
  kFinal<<<1, 1, 0, stream>>>(cnt, acc, out);
}